// m1_32736240731005
// MI455X (gfx1250) — compile-verified
//
#include <hip/hip_runtime.h>
#include <hip/hip_bf16.h>
#include <math.h>

typedef __attribute__((ext_vector_type(2))) float v2f;
typedef __attribute__((ext_vector_type(8))) float v8f;

#define HEADS 4
#define NEG_SLOPE 0.2f

// ---------- ordered-uint encoding for float atomic max ----------
__device__ __forceinline__ unsigned f2ord(float f) {
  unsigned b = __float_as_uint(f);
  return (b & 0x80000000u) ? ~b : (b | 0x80000000u);
}
__device__ __forceinline__ float ord2f(unsigned u) {
  unsigned b = (u & 0x80000000u) ? (u & 0x7FFFFFFFu) : ~u;
  return __uint_as_float(b);
}

// ---------- WMMA fp32 GEMM: [nrows,128] @ [128,16] -> [nrows,16] ----------
// One wave per 16-row tile; K=128 via 32 x V_WMMA_F32_16X16X4_F32.
__global__ void wmma_gemm_k128_n16(const float* __restrict__ X,
                                   const float* __restrict__ W,
                                   float* __restrict__ OUT, int nrows) {
  int gtid = blockIdx.x * blockDim.x + threadIdx.x;
  int wave = gtid >> 5;
  int lane = threadIdx.x & 31;
  int row0 = wave * 16;
  if (row0 >= nrows) return;
  int m  = lane & 15;          // A row / B col for this lane
  int kk = (lane >> 4) << 1;   // lanes 0-15: K=0,1 ; lanes 16-31: K=2,3
  const float* xr = X + (size_t)(row0 + m) * 128;
  v8f c = {};
#pragma unroll
  for (int k0 = 0; k0 < 128; k0 += 4) {
    v2f a = { xr[k0 + kk], xr[k0 + kk + 1] };
    v2f b = { W[(k0 + kk) * 16 + m], W[(k0 + kk + 1) * 16 + m] };
    c = __builtin_amdgcn_wmma_f32_16x16x4_f32(false, a, false, b, (short)0, c,
                                              false, false);
  }
  int mbase = (lane >> 4) * 8;   // C/D: vgpr j -> M=j (lanes 0-15) / M=j+8
  float* o = OUT + (size_t)row0 * 16;
#pragma unroll
  for (int j = 0; j < 8; ++j) o[(size_t)(mbase + j) * 16 + m] = c[j];
}

// ---------- WMMA fp32 GEMM: [nrows,16] @ [16,128] -> [nrows,128] ----------
// One wave per (16-row, 16-col) tile; 8 col tiles; K=16 via 4 WMMA ops.
__global__ void wmma_gemm_k16_n128(const float* __restrict__ X,
                                   const float* __restrict__ W,
                                   float* __restrict__ OUT, int nrows) {
  int gtid = blockIdx.x * blockDim.x + threadIdx.x;
  int wave = gtid >> 5;
  int lane = threadIdx.x & 31;
  int rt = wave >> 3;
  int ct = wave & 7;
  int row0 = rt * 16;
  if (row0 >= nrows) return;
  int m  = lane & 15;
  int kk = (lane >> 4) << 1;
  const float* xr = X + (size_t)(row0 + m) * 16;
  int nc = ct * 16 + m;
  v8f c = {};
#pragma unroll
  for (int k0 = 0; k0 < 16; k0 += 4) {
    v2f a = { xr[k0 + kk], xr[k0 + kk + 1] };
    v2f b = { W[(k0 + kk) * 128 + nc], W[(k0 + kk + 1) * 128 + nc] };
    c = __builtin_amdgcn_wmma_f32_16x16x4_f32(false, a, false, b, (short)0, c,
                                              false, false);
  }
  int mbase = (lane >> 4) * 8;
#pragma unroll
  for (int j = 0; j < 8; ++j)
    OUT[(size_t)(row0 + mbase + j) * 128 + nc] = c[j];
}

// ---------- small helpers ----------
__global__ void zero_f(float* p, long long cnt) {
  long long i = (long long)blockIdx.x * blockDim.x + threadIdx.x;
  if (i < cnt) p[i] = 0.f;
}
__global__ void fill_f(float* p, long long cnt, float v) {
  long long i = (long long)blockIdx.x * blockDim.x + threadIdx.x;
  if (i < cnt) p[i] = v;
}
__global__ void deg_edges(const int* __restrict__ ei, int E, float* __restrict__ deg) {
  int e = blockIdx.x * blockDim.x + threadIdx.x;
  if (e >= E) return;
  atomicAdd(&deg[ei[E + e]], 1.0f);
}
__global__ void rsqrt_ip(float* p, int n) {
  int i = blockIdx.x * blockDim.x + threadIdx.x;
  if (i < n) p[i] = rsqrtf(p[i]);
}
__global__ void bias_relu(float* __restrict__ p, const float* __restrict__ b,
                          long long cnt, int F) {
  long long i = (long long)blockIdx.x * blockDim.x + threadIdx.x;
  if (i >= cnt) return;
  float v = p[i] + b[i % F];
  p[i] = v > 0.f ? v : 0.f;
}

// ---------- GCN scatter: OUT[col,:16] += H[row,:16]*dis[row]*dis[col] ----------
__global__ void gcn_scatter16(const float* __restrict__ H, const int* __restrict__ ei,
                              int E, int n, const float* __restrict__ dis,
                              float* __restrict__ OUT) {
  int e = blockIdx.x * blockDim.x + threadIdx.x;
  if (e >= E + n) return;
  int r, c;
  if (e < E) { r = ei[e]; c = ei[E + e]; } else { r = c = e - E; }
  float norm = dis[r] * dis[c];
  const float4* hr = (const float4*)(H + (size_t)r * 16);
  float* o = OUT + (size_t)c * 16;
#pragma unroll
  for (int q = 0; q < 4; ++q) {
    float4 v = hr[q];
    atomicAdd(o + q * 4 + 0, v.x * norm);
    atomicAdd(o + q * 4 + 1, v.y * norm);
    atomicAdd(o + q * 4 + 2, v.z * norm);
    atomicAdd(o + q * 4 + 3, v.w * norm);
  }
}

// ---------- GAT: per-(node,head) attention dot products ----------
__global__ void attn_dots(const float* __restrict__ H2, const float* __restrict__ as,
                          const float* __restrict__ ad, float* __restrict__ osrc,
                          float* __restrict__ odst, int n) {
  int g = blockIdx.x * blockDim.x + threadIdx.x;
  if (g >= n * HEADS) return;
  int node = g >> 2, hh = g & 3;
  const float* hv = H2 + (size_t)node * 128 + hh * 32;
  const float* vs = as + hh * 32;
  const float* vd = ad + hh * 32;
  float s1 = 0.f, s2 = 0.f;
#pragma unroll
  for (int c = 0; c < 32; ++c) { float x = hv[c]; s1 += x * vs[c]; s2 += x * vd[c]; }
  osrc[g] = s1;
  odst[g] = s2;
}

__global__ void gat_emax(const float* __restrict__ asrc, const float* __restrict__ adst,
                         const int* __restrict__ ei, int E, int n,
                         unsigned* __restrict__ emaxu) {
  int e = blockIdx.x * blockDim.x + threadIdx.x;
  if (e >= E + n) return;
  int r, c;
  if (e < E) { r = ei[e]; c = ei[E + e]; } else { r = c = e - E; }
#pragma unroll
  for (int hh = 0; hh < HEADS; ++hh) {
    float v = asrc[r * HEADS + hh] + adst[c * HEADS + hh];
    v = v > 0.f ? v : v * NEG_SLOPE;
    atomicMax(&emaxu[c * HEADS + hh], f2ord(v));
  }
}

__global__ void gat_denom(const float* __restrict__ asrc, const float* __restrict__ adst,
                          const int* __restrict__ ei, int E, int n,
                          const unsigned* __restrict__ emaxu, float* __restrict__ denom) {
  int e = blockIdx.x * blockDim.x + threadIdx.x;
  if (e >= E + n) return;
  int r, c;
  if (e < E) { r = ei[e]; c = ei[E + e]; } else { r = c = e - E; }
#pragma unroll
  for (int hh = 0; hh < HEADS; ++hh) {
    float v = asrc[r * HEADS + hh] + adst[c * HEADS + hh];
    v = v > 0.f ? v : v * NEG_SLOPE;
    float em = ord2f(emaxu[c * HEADS + hh]);
    atomicAdd(&denom[c * HEADS + hh], expf(v - em));
  }
}

// 32 threads per edge; each thread owns 4 consecutive features (one head per 8 lanes)
__global__ void gat_scatter(const float* __restrict__ H2, const float* __restrict__ asrc,
                            const float* __restrict__ adst,
                            const unsigned* __restrict__ emaxu,
                            const float* __restrict__ denom, const int* __restrict__ ei,
                            int E, int n, float* __restrict__ OUT) {
  long long g = (long long)blockIdx.x * blockDim.x + threadIdx.x;
  int e = (int)(g >> 5);
  int lane = (int)(g & 31);
  if (e >= E + n) return;
  int r, c;
  if (e < E) { r = ei[e]; c = ei[E + e]; } else { r = c = e - E; }
  int hh = lane >> 3;
  float v = asrc[r * HEADS + hh] + adst[c * HEADS + hh];
  v = v > 0.f ? v : v * NEG_SLOPE;
  float em = ord2f(emaxu[c * HEADS + hh]);
  float alpha = expf(v - em) / (denom[c * HEADS + hh] + 1e-16f);
  float4 hv = ((const float4*)(H2 + (size_t)r * 128))[lane];
  float* o = OUT + (size_t)c * 128 + lane * 4;
  atomicAdd(o + 0, hv.x * alpha);
  atomicAdd(o + 1, hv.y * alpha);
  atomicAdd(o + 2, hv.z * alpha);
  atomicAdd(o + 3, hv.w * alpha);
}

// ---------- final bias + log_softmax over 16 classes ----------
__global__ void log_softmax16(const float* __restrict__ acc, const float* __restrict__ b3,
                              float* __restrict__ out, int n) {
  int i = blockIdx.x * blockDim.x + threadIdx.x;
  if (i >= n) return;
  const float* a = acc + (size_t)i * 16;
  float s[16];
  float m = -3.0e38f;
#pragma unroll
  for (int j = 0; j < 16; ++j) { s[j] = a[j] + b3[j]; m = fmaxf(m, s[j]); }
  float sum = 0.f;
#pragma unroll
  for (int j = 0; j < 16; ++j) sum += expf(s[j] - m);
  float l = logf(sum);
  float* o = out + (size_t)i * 16;
#pragma unroll
  for (int j = 0; j < 16; ++j) o[j] = s[j] - m - l;
}

static inline int nblk(long long n, int b) { return (int)((n + b - 1) / b); }

extern "C" void kernel_launch(void* const* d_in, const int* in_sizes, int n_in,
                              void* d_out, int out_size, void* d_ws, size_t ws_size,
                              hipStream_t stream) {
  const float* x       = (const float*)d_in[0];
  const int*   ei      = (const int*)d_in[1];
  const float* W1      = (const float*)d_in[2];
  const float* b1      = (const float*)d_in[3];
  const float* Wg      = (const float*)d_in[4];
  const float* att_src = (const float*)d_in[5];
  const float* att_dst = (const float*)d_in[6];
  const float* bg      = (const float*)d_in[7];
  const float* W3      = (const float*)d_in[8];
  const float* b3      = (const float*)d_in[9];
  float* out = (float*)d_out;

  const int N  = in_sizes[0] / 128;
  const int E  = in_sizes[1] / 2;
  const int EP = E + N;

  // workspace layout (floats)
  float* ws = (float*)d_ws;
  float*    h     = ws;                          // 16N  (GEMM1 out; later h4)
  float*    dis   = ws + 16  * (size_t)N;        // N
  float*    h1    = ws + 17  * (size_t)N;        // 16N
  float*    h2    = ws + 33  * (size_t)N;        // 128N (GEMM2 out)
  float*    outac = h2;                          // alias: h2 dead before use
  float*    asrc  = ws + 161 * (size_t)N;        // 4N
  float*    adst  = ws + 165 * (size_t)N;        // 4N
  unsigned* emaxu = (unsigned*)(ws + 169 * (size_t)N); // 4N
  float*    denom = ws + 173 * (size_t)N;        // 4N
  float*    h3    = ws + 177 * (size_t)N;        // 128N

  const int B = 256;
  int tiles = (N + 15) / 16;

  // 1) h = x @ W1   (WMMA fp32)
  wmma_gemm_k128_n16<<<nblk((long long)tiles * 32, B), B, 0, stream>>>(x, W1, h, N);

  // 2) degree with self-loops -> dis = deg^-1/2
  fill_f<<<nblk(N, B), B, 0, stream>>>(dis, N, 1.0f);
  deg_edges<<<nblk(E, B), B, 0, stream>>>(ei, E, dis);
  rsqrt_ip<<<nblk(N, B), B, 0, stream>>>(dis, N);

  // 3) GCN1 scatter + bias + relu -> h1
  zero_f<<<nblk(16LL * N, B), B, 0, stream>>>(h1, 16LL * N);
  gcn_scatter16<<<nblk(EP, B), B, 0, stream>>>(h, ei, E, N, dis, h1);
  bias_relu<<<nblk(16LL * N, B), B, 0, stream>>>(h1, b1, 16LL * N, 16);

  // 4) h2 = h1 @ Wg  (WMMA fp32)
  wmma_gemm_k16_n128<<<nblk((long long)tiles * 8 * 32, B), B, 0, stream>>>(h1, Wg, h2, N);

  // 5) attention scalars
  attn_dots<<<nblk((long long)N * HEADS, B), B, 0, stream>>>(h2, att_src, att_dst,
                                                             asrc, adst, N);

  // 6) segment softmax (max, sum) over destination nodes
  zero_f<<<nblk(4LL * N, B), B, 0, stream>>>((float*)emaxu, 4LL * N);
  gat_emax<<<nblk(EP, B), B, 0, stream>>>(asrc, adst, ei, E, N, emaxu);
  zero_f<<<nblk(4LL * N, B), B, 0, stream>>>(denom, 4LL * N);
  gat_denom<<<nblk(EP, B), B, 0, stream>>>(asrc, adst, ei, E, N, emaxu, denom);

  // 7) weighted message scatter -> h3; bias + relu
  zero_f<<<nblk(128LL * N, B), B, 0, stream>>>(h3, 128LL * N);
  gat_scatter<<<nblk((long long)EP * 32, B), B, 0, stream>>>(h2, asrc, adst, emaxu,
                                                             denom, ei, E, N, h3);
  bias_relu<<<nblk(128LL * N, B), B, 0, stream>>>(h3, bg, 128LL * N, 128);

  // 8) h4 = h3 @ W3 (into h buffer, WMMA fp32)
  wmma_gemm_k128_n16<<<nblk((long long)tiles * 32, B), B, 0, stream>>>(h3, W3, h, N);

  // 9) GCN2 scatter -> outac ; log_softmax -> d_out
  zero_f<<<nblk(16LL * N, B), B, 0, stream>>>(outac, 16LL * N);
  gcn_scatter16<<<nblk(EP, B), B, 0, stream>>>(h, ei, E, N, dis, outac);
  log_softmax16<<<nblk(N, B), B, 0, stream>>>(outac, b3, out, N);
}